// NeuralMemory_25632364823053
// MI455X (gfx1250) — compile-verified
//
#include <hip/hip_runtime.h>

#define T_DIM 128
#define B_DIM 128
#define R_DIM 512
#define NSLOTS 256          // 2*T
#define SPAD   257          // LDS row stride (bank-conflict-free both ways)

typedef float v2f __attribute__((ext_vector_type(2)));
typedef float v8f __attribute__((ext_vector_type(8)));

// ---------------------------------------------------------------------------
// Kernel 1: sequential weight computation (the scan part of the reference).
// One block, 128 threads, thread b owns batch b.
//   w[t][b][j] = min(S_old[j][b], scal_j),  scal_j = max_b (u_t[b] - used_j[b])
//   pop: S[j][b] -= min(S_old, max(0, u_t[b] - used_j[b]))
// used_j[b] = suffix sum of S over slots processed before j (newest->oldest).
// ---------------------------------------------------------------------------
__global__ __launch_bounds__(128) void weights_kernel(
    const float* __restrict__ u,    // (T,B)
    const float* __restrict__ d1,   // (T,B)
    const float* __restrict__ d2,   // (T,B)
    float* __restrict__ W)          // (T,B,NSLOTS)
{
    __shared__ float S[B_DIM * SPAD];     // stack depths, row per batch
    __shared__ float X[B_DIM * SPAD];     // u_t - used_j, row per batch
    __shared__ float scal[NSLOTS];        // batch-wide max per slot

    const int b = threadIdx.x;            // 0..127
    float* Srow = &S[b * SPAD];
    float* Xrow = &X[b * SPAD];

    #pragma unroll 8
    for (int j = 0; j < NSLOTS; ++j) Srow[j] = 0.0f;

    for (int t = 0; t < T_DIM; ++t) {
        const int jmax = 2 * t + 1;
        const float ut = u[t * B_DIM + b];
        // push the two new slots
        Srow[2 * t]     = d1[t * B_DIM + b];
        Srow[2 * t + 1] = d2[t * B_DIM + b];

        // Phase A: per-batch suffix scan, X[b][j] = ut - used (before adding S[j])
        float used = 0.0f;
        for (int j = jmax; j >= 0; --j) {
            Xrow[j] = ut - used;
            used += Srow[j];
        }
        __syncthreads();   // X complete before cross-batch reduction

        // Phase B: scal[j] = max over batches of X[b][j] (transposed, conflict-free)
        for (int jj = threadIdx.x; jj <= jmax; jj += B_DIM) {
            float m = -INFINITY;
            #pragma unroll 8
            for (int bb = 0; bb < B_DIM; ++bb)
                m = fmaxf(m, X[bb * SPAD + jj]);
            scal[jj] = m;
        }
        __syncthreads();   // scal ready; also protects X vs next Phase A

        // Phase C: emit weights, apply pop
        float* wrow = W + ((size_t)t * B_DIM + b) * NSLOTS;
        for (int j = 0; j < NSLOTS; ++j) {
            float w = 0.0f;
            if (j <= jmax) {
                const float s = Srow[j];
                w = fminf(s, scal[j]);                       // may be negative (scal)
                const float take = fminf(s, fmaxf(0.0f, Xrow[j]));
                Srow[j] = s - take;
            }
            wrow[j] = w;
        }
        // no barrier needed: next-iter A touches only own rows; B(t+1) barrier covers rest
    }
}

// ---------------------------------------------------------------------------
// Kernel 2: batched GEMM  out[b] = W[b] (128x256) @ Vm[b] (256x512), fp32 WMMA.
// Vm[2t'] = v1[t'], Vm[2t'+1] = v2[t'].
// Wave owns a 16(M) x 64(N) tile of one batch: 4 accumulators, K-loop of 64
// v_wmma_f32_16x16x4_f32 steps per accumulator chain.
// A 16x4 f32 layout: lane = m (0-15) | m (16-31 -> K=2,3); vgpr k parity.
// B 4x16 f32 layout: lane = n, lanes 16-31 hold K rows 2,3.
// D 16x16 f32: vgpr r -> M = r + 8*(lane/16), N = lane%16.
// ---------------------------------------------------------------------------
__global__ __launch_bounds__(256) void gemm_kernel(
    const float* __restrict__ W,    // (T,B,NSLOTS)
    const float* __restrict__ v1,   // (T,B,R)
    const float* __restrict__ v2,   // (T,B,R)
    float* __restrict__ out)        // (T,B,R)
{
    const int lane = threadIdx.x & 31;
    const int wave = threadIdx.x >> 5;      // 0..7 -> N-group
    const int lm   = lane & 15;
    const int half = lane >> 4;             // 0: K=0,1   1: K=2,3

    const int b  = blockIdx.x >> 3;         // 0..127
    const int mt = blockIdx.x & 7;          // 0..7 (t-tile)
    const int n0 = wave * 64;               // N-group base

    v8f acc0 = {}, acc1 = {}, acc2 = {}, acc3 = {};

    // A: row t = mt*16 + lm of W[b], columns k + 2*half + {0,1}
    const float* Ap = W + ((size_t)(mt * 16 + lm) * B_DIM + b) * NSLOTS + 2 * half;
    // B: element (kk, n) = v1/v2[(kk>>1)][b][n]; kk = k + 2*half (even -> v1, +1 -> v2)
    const size_t bstride = (size_t)B_DIM * R_DIM;
    const float* B1 = v1 + (size_t)half * bstride + (size_t)b * R_DIM + n0 + lm;
    const float* B2 = v2 + (size_t)half * bstride + (size_t)b * R_DIM + n0 + lm;

    #pragma unroll 4
    for (int k = 0; k < NSLOTS; k += 4) {
        const v2f a = *(const v2f*)(Ap + k);
        const size_t off = (size_t)(k >> 1) * bstride;
        v2f b0, b1, b2, b3;
        b0.x = B1[off +  0]; b0.y = B2[off +  0];
        b1.x = B1[off + 16]; b1.y = B2[off + 16];
        b2.x = B1[off + 32]; b2.y = B2[off + 32];
        b3.x = B1[off + 48]; b3.y = B2[off + 48];
        acc0 = __builtin_amdgcn_wmma_f32_16x16x4_f32(false, a, false, b0, (short)0, acc0, false, false);
        acc1 = __builtin_amdgcn_wmma_f32_16x16x4_f32(false, a, false, b1, (short)0, acc1, false, false);
        acc2 = __builtin_amdgcn_wmma_f32_16x16x4_f32(false, a, false, b2, (short)0, acc2, false, false);
        acc3 = __builtin_amdgcn_wmma_f32_16x16x4_f32(false, a, false, b3, (short)0, acc3, false, false);
    }

    // Store D tiles: element (m = r + 8*half, n = lm) of each 16x16 tile
    #pragma unroll
    for (int r = 0; r < 8; ++r) {
        const int trow = mt * 16 + r + 8 * half;
        float* orow = out + ((size_t)trow * B_DIM + b) * R_DIM + n0 + lm;
        orow[ 0] = acc0[r];
        orow[16] = acc1[r];
        orow[32] = acc2[r];
        orow[48] = acc3[r];
    }
}

// ---------------------------------------------------------------------------
extern "C" void kernel_launch(void* const* d_in, const int* in_sizes, int n_in,
                              void* d_out, int out_size, void* d_ws, size_t ws_size,
                              hipStream_t stream) {
    const float* u  = (const float*)d_in[0];   // (T,B,1)
    const float* d1 = (const float*)d_in[1];   // (T,B,1)
    const float* d2 = (const float*)d_in[2];   // (T,B,1)
    const float* v1 = (const float*)d_in[3];   // (T,B,R)
    const float* v2 = (const float*)d_in[4];   // (T,B,R)
    float* out = (float*)d_out;                // (T,B,R)
    float* W   = (float*)d_ws;                 // needs T*B*NSLOTS*4 = 16 MiB

    weights_kernel<<<dim3(1), dim3(128), 0, stream>>>(u, d1, d2, W);
    gemm_kernel<<<dim3(B_DIM * 8), dim3(256), 0, stream>>>(W, v1, v2, out);
}